// GCN_44306882626196
// MI455X (gfx1250) — compile-verified
//
#include <hip/hip_runtime.h>

typedef __attribute__((ext_vector_type(2))) float v2f;
typedef __attribute__((ext_vector_type(8))) float v8f;

#define KDIM 256   // D_FEAT
#define NDIM 256   // UNITS

// ---------------------------------------------------------------------------
// Kernel 1: h = X @ W with V_WMMA_F32_16X16X4_F32 (exact f32, memory-bound).
// Block = 256 threads = 8 waves. Each wave computes a 32x32 patch of H:
// 2 M-tiles x 2 N-tiles (4 v8f accumulators). K loop steps by 4.
// A-fragment (16x4 f32): lanes 0-15 carry K=k,k+1 ; lanes 16-31 carry K=k+2,k+3
// for row M = mBase + (lane&15). B-fragment mirrored over K rows of W.
// C layout: VGPR v -> row (v + 8*(lane>>4)), col (lane&15).
// ---------------------------------------------------------------------------
__global__ __launch_bounds__(256) void gcn_gemm_wmma_f32(
    const float* __restrict__ X, const float* __restrict__ W,
    float* __restrict__ H, int M)
{
  const int tid  = threadIdx.x;
  const int lane = tid & 31;
  const int wave = tid >> 5;        // 0..7
  const int l16  = lane & 15;
  const int half = lane >> 4;       // 0 or 1

  const int mBase = blockIdx.x * 32;           // 100000 % 32 == 0 -> no tail
  const int n0 = wave * 32;                    // 8 waves cover all 256 cols
  const int n1 = n0 + 16;

  v8f acc00 = {}, acc01 = {}, acc10 = {}, acc11 = {};

  const float* a0p = X + (size_t)(mBase + l16) * KDIM;        // M-tile 0 row
  const float* a1p = a0p + (size_t)16 * KDIM;                 // M-tile 1 row

  #pragma unroll 4
  for (int k = 0; k < KDIM; k += 4) {
    const int ka = k + 2 * half;
    v2f a0 = *(const v2f*)(a0p + ka);          // X[m][ka], X[m][ka+1]
    v2f a1 = *(const v2f*)(a1p + ka);
    const float* bp = W + (size_t)ka * NDIM + l16;
    v2f b0, b1;
    b0.x = bp[n0];        b0.y = bp[n0 + NDIM];   // W[ka][n], W[ka+1][n]
    b1.x = bp[n1];        b1.y = bp[n1 + NDIM];
    acc00 = __builtin_amdgcn_wmma_f32_16x16x4_f32(false, a0, false, b0,
                                                  (short)0, acc00, false, false);
    acc10 = __builtin_amdgcn_wmma_f32_16x16x4_f32(false, a1, false, b0,
                                                  (short)0, acc10, false, false);
    acc01 = __builtin_amdgcn_wmma_f32_16x16x4_f32(false, a0, false, b1,
                                                  (short)0, acc01, false, false);
    acc11 = __builtin_amdgcn_wmma_f32_16x16x4_f32(false, a1, false, b1,
                                                  (short)0, acc11, false, false);
  }

  // Store: VGPR v of a 16x16 f32 C tile is row (v + 8*half), col l16.
  float* c00 = H + (size_t)(mBase + 8 * half) * NDIM + n0 + l16;
  #pragma unroll
  for (int v = 0; v < 8; ++v) {
    c00[(size_t)v * NDIM]             = acc00[v];
    c00[(size_t)v * NDIM + 16]        = acc01[v];
    c00[(size_t)(v + 16) * NDIM]      = acc10[v];
    c00[(size_t)(v + 16) * NDIM + 16] = acc11[v];
  }
}

// ---------------------------------------------------------------------------
// Kernel 2: out[i,j] = b[j]  (bias pre-load; atomics then accumulate on top)
// ---------------------------------------------------------------------------
__global__ __launch_bounds__(256) void gcn_init_bias(
    float* __restrict__ out, const float* __restrict__ b, int total)
{
  int i = blockIdx.x * 256 + threadIdx.x;
  if (i < total) out[i] = b[i & (NDIM - 1)];
}

// ---------------------------------------------------------------------------
// Kernel 3: COO scatter: out[row[e]] += val[e] * h[col[e]].
// One wave per edge; lane handles 8 contiguous floats (32*8 = 256) via two
// float4 gathers (L2-resident: h is 102 MB < 192 MB L2) and 8 f32 atomics.
// ---------------------------------------------------------------------------
__global__ __launch_bounds__(256) void gcn_spmm_scatter(
    const int* __restrict__ erow, const int* __restrict__ ecol,
    const float* __restrict__ eval_, const float* __restrict__ H,
    float* __restrict__ out, int nEdges)
{
  const int lane = threadIdx.x & 31;
  const int e = blockIdx.x * 8 + (threadIdx.x >> 5);
  if (e >= nEdges) return;

  const int   r = erow[e];
  const int   c = ecol[e];
  const float v = eval_[e];

  const float* hs = H + (size_t)c * NDIM + lane * 8;
  float*       op = out + (size_t)r * NDIM + lane * 8;

  float4 h0 = *(const float4*)(hs);
  float4 h1 = *(const float4*)(hs + 4);

  unsafeAtomicAdd(op + 0, v * h0.x);
  unsafeAtomicAdd(op + 1, v * h0.y);
  unsafeAtomicAdd(op + 2, v * h0.z);
  unsafeAtomicAdd(op + 3, v * h0.w);
  unsafeAtomicAdd(op + 4, v * h1.x);
  unsafeAtomicAdd(op + 5, v * h1.y);
  unsafeAtomicAdd(op + 6, v * h1.z);
  unsafeAtomicAdd(op + 7, v * h1.w);
}

// ---------------------------------------------------------------------------
// Kernel 4: in-place ReLU
// ---------------------------------------------------------------------------
__global__ __launch_bounds__(256) void gcn_relu(float* __restrict__ out, int total)
{
  int i = blockIdx.x * 256 + threadIdx.x;
  if (i < total) {
    float x = out[i];
    out[i] = x > 0.0f ? x : 0.0f;
  }
}

extern "C" void kernel_launch(void* const* d_in, const int* in_sizes, int n_in,
                              void* d_out, int out_size, void* d_ws, size_t ws_size,
                              hipStream_t stream) {
  const float* X     = (const float*)d_in[0];
  const int*   erow  = (const int*)d_in[1];
  const int*   ecol  = (const int*)d_in[2];
  const float* eval_ = (const float*)d_in[3];
  const float* W     = (const float*)d_in[4];
  const float* b     = (const float*)d_in[5];
  float* out = (float*)d_out;
  float* H   = (float*)d_ws;                 // [M, 256] f32 scratch (102.4 MB)

  const int M = in_sizes[0] / KDIM;          // 100000
  const int E = in_sizes[1];                 // 3200000
  const int total = M * NDIM;

  dim3 blk(256);
  gcn_gemm_wmma_f32<<<dim3((M + 31) / 32), blk, 0, stream>>>(X, W, H, M);
  gcn_init_bias<<<dim3((total + 255) / 256), blk, 0, stream>>>(out, b, total);
  gcn_spmm_scatter<<<dim3((E + 7) / 8), blk, 0, stream>>>(erow, ecol, eval_, H, out, E);
  gcn_relu<<<dim3((total + 255) / 256), blk, 0, stream>>>(out, total);
}